// GAT_61976378081726
// MI455X (gfx1250) — compile-verified
//
#include <hip/hip_runtime.h>
#include <math.h>

#define F 20
#define NEG_SLOPE 0.2f

typedef float v2f __attribute__((ext_vector_type(2)));
typedef float v8f __attribute__((ext_vector_type(8)));

// ---------------------------------------------------------------------------
// Float atomic-max via monotone ordered-int mapping (no CAS loop).
// m[] must be initialized to -FLT_MAX. Positive candidates go through signed
// atomicMax (positive float bits are ordered as signed ints); negative
// candidates through unsigned atomicMin (negative float bits reverse-ordered
// as unsigned). Mixing the two ops preserves "stored value only increases in
// float order", so the final value is the true max.
// ---------------------------------------------------------------------------
__device__ __forceinline__ void atomicMaxF32(float* addr, float val) {
    if (val >= 0.0f) {
        atomicMax((int*)addr, __float_as_int(val));
    } else {
        atomicMin((unsigned int*)addr, __float_as_uint(val));
    }
}

// ---------------------------------------------------------------------------
// Layer 1 node pass: Fin == 1, so h[i,f] = x[i] * W1[f]. Also compute the
// per-node attention scalars and reset the per-layer accumulators.
// ---------------------------------------------------------------------------
__global__ void gat_node_init_l1(const float* __restrict__ x,
                                 const float* __restrict__ W1,
                                 const float* __restrict__ a_src,
                                 const float* __restrict__ a_dst,
                                 float* __restrict__ h,
                                 float* __restrict__ asrc,
                                 float* __restrict__ adst,
                                 float* __restrict__ m,
                                 float* __restrict__ denom,
                                 float* __restrict__ accum,
                                 int N) {
    int i = blockIdx.x * blockDim.x + threadIdx.x;
    if (i >= N) return;
    float xv = x[i];
    float sa = 0.0f, sd = 0.0f;
#pragma unroll
    for (int f = 0; f < F; ++f) {
        float hv = xv * W1[f];
        h[i * F + f] = hv;
        sa += hv * a_src[f];
        sd += hv * a_dst[f];
        accum[i * F + f] = 0.0f;
    }
    asrc[i] = sa;
    adst[i] = sd;
    m[i] = -3.402823466e38f;
    denom[i] = 0.0f;
}

// ---------------------------------------------------------------------------
// Layer 2 GEMM: h = feat [N x 20] @ W [20 x 20] using V_WMMA_F32_16X16X4_F32.
// One wave (32 lanes) per 16-row tile; EXEC stays all-ones through the entire
// WMMA loop: the zero-padding of output columns 20..31 is done branch-free
// with a clamped (always in-bounds) column index and a 0/1 lane mask, so the
// padded B loads stay plain clause-able global_load_b32 (no exec save/restore).
// K = 20 = 5 steps of K=4. N = 20 -> two 16-wide column tiles (second padded).
// A layout (ISA 7.12.2, 32-bit 16x4): lanes 0-15 hold K0/K1, lanes 16-31 K2/K3.
// C layout: VGPR j = row (j + 8*(lane>=16)), lane&15 = column.
// ---------------------------------------------------------------------------
__global__ void __launch_bounds__(32)
gat_wmma_feat_x_w(const float* __restrict__ feat,
                  const float* __restrict__ W,   // [20 x 20], row-major (K x Nout)
                  float* __restrict__ h,         // [N x 20]
                  int N) {
    int tile  = blockIdx.x;
    int lane  = threadIdx.x;
    int mrow  = lane & 15;
    int khalf = (lane >> 4) * 2;            // 0 for lanes 0-15, 2 for 16-31

    int row  = tile * 16 + mrow;
    int rowc = row < N ? row : N - 1;       // clamp (branch-free; keeps EXEC full)
    const float* arow = feat + (size_t)rowc * F;

    // Second N-tile: columns 16..31, valid only while < 20. Branch-free pad.
    int   n1   = 16 + mrow;
    int   n1c  = (n1 < F) ? n1 : 0;         // clamped, always in-bounds
    float msk1 = (n1 < F) ? 1.0f : 0.0f;    // lane mask for padding

    v8f c0 = {};
    v8f c1 = {};
#pragma unroll
    for (int k0 = 0; k0 < F; k0 += 4) {
        const float* wr0 = W + (size_t)(k0 + khalf + 0) * F;
        const float* wr1 = W + (size_t)(k0 + khalf + 1) * F;

        v2f a;
        a.x = arow[k0 + khalf + 0];
        a.y = arow[k0 + khalf + 1];

        // B tile 0: output columns 0..15 (always valid)
        v2f b0;
        b0.x = wr0[mrow];
        b0.y = wr1[mrow];
        c0 = __builtin_amdgcn_wmma_f32_16x16x4_f32(false, a, false, b0,
                                                   (short)0, c0, false, false);

        // B tile 1: output columns 16..19, rest zero via lane mask
        v2f b1;
        b1.x = wr0[n1c] * msk1;
        b1.y = wr1[n1c] * msk1;
        c1 = __builtin_amdgcn_wmma_f32_16x16x4_f32(false, a, false, b1,
                                                   (short)0, c1, false, false);
    }

    // Write back D: lane L, VGPR j -> row tile*16 + j + 8*(L>=16), col (L&15)+16*ntile
    int rbase = tile * 16 + ((lane >> 4) << 3);
#pragma unroll
    for (int j = 0; j < 8; ++j) {
        int r = rbase + j;
        if (r < N) {
            h[(size_t)r * F + mrow] = c0[j];
            if (n1 < F) h[(size_t)r * F + n1] = c1[j];
        }
    }
}

// ---------------------------------------------------------------------------
// Node pass (layer 2 variant): attention scalars from h, reset accumulators.
// ---------------------------------------------------------------------------
__global__ void gat_node_alpha_init(const float* __restrict__ h,
                                    const float* __restrict__ a_src,
                                    const float* __restrict__ a_dst,
                                    float* __restrict__ asrc,
                                    float* __restrict__ adst,
                                    float* __restrict__ m,
                                    float* __restrict__ denom,
                                    float* __restrict__ accum,
                                    int N) {
    int i = blockIdx.x * blockDim.x + threadIdx.x;
    if (i >= N) return;
    float sa = 0.0f, sd = 0.0f;
#pragma unroll
    for (int f = 0; f < F; ++f) {
        float hv = h[i * F + f];
        sa += hv * a_src[f];
        sd += hv * a_dst[f];
        accum[i * F + f] = 0.0f;
    }
    asrc[i] = sa;
    adst[i] = sd;
    m[i] = -3.402823466e38f;
    denom[i] = 0.0f;
}

// ---------------------------------------------------------------------------
// Edge pass 1: segment max of leaky_relu(asrc[s] + adst[d]) over dst.
// Edges e in [0,E) come from edge_index; e in [E, E+N) are the self-loops.
// ---------------------------------------------------------------------------
__global__ void gat_edge_max(const int* __restrict__ src,
                             const int* __restrict__ dst,
                             const float* __restrict__ asrc,
                             const float* __restrict__ adst,
                             float* __restrict__ m,
                             int E, int Etot) {
    int e = blockIdx.x * blockDim.x + threadIdx.x;
    if (e >= Etot) return;
    int s, d;
    if (e < E) { s = src[e]; d = dst[e]; }
    else       { s = e - E; d = e - E; }
    float ev = asrc[s] + adst[d];
    ev = ev > 0.0f ? ev : NEG_SLOPE * ev;
    atomicMaxF32(&m[d], ev);
}

// ---------------------------------------------------------------------------
// Edge pass 2: t = exp(e - m[d]); denom[d] += t; accum[d,:] += t * h[s,:].
// The softmax division is deferred to the per-node finish kernel, which
// removes the third edge pass entirely. All scatter work is native
// global_atomic_add_f32 resolved in the 192 MB L2 (node arrays ~34 MB).
// ---------------------------------------------------------------------------
__global__ void gat_edge_acc(const int* __restrict__ src,
                             const int* __restrict__ dst,
                             const float* __restrict__ asrc,
                             const float* __restrict__ adst,
                             const float* __restrict__ m,
                             const float* __restrict__ h,
                             float* __restrict__ denom,
                             float* __restrict__ accum,
                             int E, int Etot) {
    int e = blockIdx.x * blockDim.x + threadIdx.x;
    if (e >= Etot) return;
    int s, d;
    if (e < E) { s = src[e]; d = dst[e]; }
    else       { s = e - E; d = e - E; }
    float ev = asrc[s] + adst[d];
    ev = ev > 0.0f ? ev : NEG_SLOPE * ev;
    float t = __expf(ev - m[d]);
    atomicAdd(&denom[d], t);
    const float* hs = h + (size_t)s * F;
    float* ac = accum + (size_t)d * F;
#pragma unroll
    for (int f = 0; f < F; ++f) {
        atomicAdd(&ac[f], t * hs[f]);
    }
}

// ---------------------------------------------------------------------------
// Node finish: feat[i,:] = relu(accum[i,:]/(denom[i]+1e-16) + b)
// ---------------------------------------------------------------------------
__global__ void gat_node_finish(const float* __restrict__ accum,
                                const float* __restrict__ denom,
                                const float* __restrict__ b,
                                float* __restrict__ feat,
                                int N) {
    int i = blockIdx.x * blockDim.x + threadIdx.x;
    if (i >= N) return;
    float inv = 1.0f / (denom[i] + 1e-16f);
#pragma unroll
    for (int f = 0; f < F; ++f) {
        float v = accum[i * F + f] * inv + b[f];
        feat[i * F + f] = v > 0.0f ? v : 0.0f;
    }
}

// ---------------------------------------------------------------------------
// Final linear: out[i] = dot(feat[i,:], Wl) + bl
// ---------------------------------------------------------------------------
__global__ void gat_final_linear(const float* __restrict__ feat,
                                 const float* __restrict__ Wl,
                                 const float* __restrict__ bl,
                                 float* __restrict__ out,
                                 int N) {
    int i = blockIdx.x * blockDim.x + threadIdx.x;
    if (i >= N) return;
    float sacc = bl[0];
#pragma unroll
    for (int f = 0; f < F; ++f) sacc += feat[i * F + f] * Wl[f];
    out[i] = sacc;
}

extern "C" void kernel_launch(void* const* d_in, const int* in_sizes, int n_in,
                              void* d_out, int out_size, void* d_ws, size_t ws_size,
                              hipStream_t stream) {
    (void)n_in; (void)out_size; (void)ws_size;

    const int N    = in_sizes[0];          // x is [N,1]
    const int E    = in_sizes[1] / 2;      // edge_index is [2,E]
    const int Etot = E + N;                // + self loops

    const float* x    = (const float*)d_in[0];
    const int*   ei   = (const int*)d_in[1];
    const float* W1   = (const float*)d_in[2];
    const float* as1  = (const float*)d_in[3];
    const float* ad1  = (const float*)d_in[4];
    const float* b1   = (const float*)d_in[5];
    const float* W2   = (const float*)d_in[6];
    const float* as2  = (const float*)d_in[7];
    const float* ad2  = (const float*)d_in[8];
    const float* b2   = (const float*)d_in[9];
    const float* Wl   = (const float*)d_in[10];
    const float* bl   = (const float*)d_in[11];

    const int* srcIdx = ei;        // row 0
    const int* dstIdx = ei + E;    // row 1

    // Workspace carving (floats): 84*N total ~= 33.6 MB for N = 100K.
    float* ws     = (float*)d_ws;
    float* h      = ws + (size_t)0  * N;   // [N x 20] transformed features
    float* accum  = ws + (size_t)20 * N;   // [N x 20] weighted sums
    float* feat1  = ws + (size_t)40 * N;   // [N x 20] layer-1 output
    float* feat2  = ws + (size_t)60 * N;   // [N x 20] layer-2 output
    float* asrc   = ws + (size_t)80 * N;   // [N]
    float* adst   = ws + (size_t)81 * N;   // [N]
    float* m      = ws + (size_t)82 * N;   // [N] segment max
    float* denom  = ws + (size_t)83 * N;   // [N] segment sum

    const int TB   = 256;
    const int gN   = (N + TB - 1) / TB;
    const int gE   = (Etot + TB - 1) / TB;
    const int gMM  = (N + 15) / 16;        // one wave per 16-row WMMA tile

    // ---- Layer 1 ----
    gat_node_init_l1<<<gN, TB, 0, stream>>>(x, W1, as1, ad1,
                                            h, asrc, adst, m, denom, accum, N);
    gat_edge_max<<<gE, TB, 0, stream>>>(srcIdx, dstIdx, asrc, adst, m, E, Etot);
    gat_edge_acc<<<gE, TB, 0, stream>>>(srcIdx, dstIdx, asrc, adst, m, h,
                                        denom, accum, E, Etot);
    gat_node_finish<<<gN, TB, 0, stream>>>(accum, denom, b1, feat1, N);

    // ---- Layer 2 (GEMM via FP32 WMMA) ----
    gat_wmma_feat_x_w<<<gMM, 32, 0, stream>>>(feat1, W2, h, N);
    gat_node_alpha_init<<<gN, TB, 0, stream>>>(h, as2, ad2,
                                               asrc, adst, m, denom, accum, N);
    gat_edge_max<<<gE, TB, 0, stream>>>(srcIdx, dstIdx, asrc, adst, m, E, Etot);
    gat_edge_acc<<<gE, TB, 0, stream>>>(srcIdx, dstIdx, asrc, adst, m, h,
                                        denom, accum, E, Etot);
    gat_node_finish<<<gN, TB, 0, stream>>>(accum, denom, b2, feat2, N);

    // ---- Final linear head ----
    gat_final_linear<<<gN, TB, 0, stream>>>(feat2, Wl, bl, (float*)d_out, N);
}